// ProposalLayer_49340584296963
// MI455X (gfx1250) — compile-verified
//
#include <hip/hip_runtime.h>
#include <hip/hip_bf16.h>
#include <stdint.h>

typedef unsigned int u32;
typedef unsigned long long u64;
typedef unsigned int v4u __attribute__((ext_vector_type(4)));
typedef int v4i __attribute__((ext_vector_type(4)));
typedef int v8i __attribute__((ext_vector_type(8)));

#define N_TOT    147456      // 128*128*9 proposals
#define ANC_A    9
#define DIM_H    128
#define DIM_W    128
#define TOPN     12000
#define POSTN    1000
#define NBINS    4096
#define CAND_CAP 32768
#define NWORDS   375         // TOPN / 32 exactly
#define NCHUNK   47          // ceil(375/8) column-block chunks
#define IOU_THR  0.7f
#define MIN_FRAC 0.015f

__device__ __forceinline__ u32 flip_f32(float x) {
  u32 u = __float_as_uint(x);
  return (u & 0x80000000u) ? ~u : (u | 0x80000000u);
}
__device__ __forceinline__ int score_bin(float s) {
  if (!(s >= 0.0f)) return 0;                 // negatives / -inf / NaN -> bin 0
  int b = (int)(s * 4096.0f);
  return b > (NBINS - 1) ? (NBINS - 1) : b;
}

// TDM 1-D tile load: 1024 f32 (4KB) from global to LDS. Descriptor per ISA 8.3/8.4.
__device__ __forceinline__ void tdm_load_4kb(u32 lds_addr, u64 gaddr) {
  v4u g0;
  g0.x = 1u;                                   // count=1, user mode, no gather
  g0.y = lds_addr;                             // bits 63:32  lds_addr
  g0.z = (u32)gaddr;                           // bits 95:64  global_addr[31:0]
  g0.w = (u32)(gaddr >> 32) | (2u << 30);      // global_addr[56:32] | type=2
  v8i g1;
  g1[0] = (int)(2u << 16);                     // data_size=2 (4 bytes), mask=0
  g1[1] = (int)(1024u << 16);                  // tensor_dim0[15:0]=1024 (atomic_addr=0)
  g1[2] = (int)(1u << 16);                     // tensor_dim0[31:16]=0, tensor_dim1=1
  g1[3] = (int)(1024u << 16);                  // tile_dim0=1024
  g1[4] = 0;                                   // tile_dim1=0, tile_dim2=0 (1-D tile)
  g1[5] = 1024;                                // tensor_dim0_stride[31:0]
  g1[6] = 0;
  g1[7] = 0;
  v4i z4 = {0, 0, 0, 0};
#if __has_include(<hip/amd_detail/amd_gfx1250_TDM.h>)
  v8i z8 = {0, 0, 0, 0, 0, 0, 0, 0};
  __builtin_amdgcn_tensor_load_to_lds(g0, g1, z4, z4, z8, 0);   // clang-23 lane: 6 args
#else
  __builtin_amdgcn_tensor_load_to_lds(g0, g1, z4, z4, 0);       // ROCm 7.2 lane: 5 args
#endif
}

// ---------------------------------------------------------------- utilities
__global__ void k_zero(u32* p, int n) {
  int i = blockIdx.x * blockDim.x + threadIdx.x;
  if (i < n) p[i] = 0u;
}

// ---------------------------------------------------------------- 1) decode
__global__ void k_decode(const float* __restrict__ cls, const float* __restrict__ loc,
                         const float* __restrict__ anc, const float* __restrict__ ii,
                         float* __restrict__ boxes, float* __restrict__ scores,
                         u32* __restrict__ hist) {
  int i = blockIdx.x * blockDim.x + threadIdx.x;
  if (i >= N_TOT) return;
  int a = i % ANC_A; int hw = i / ANC_A;
  int w = hw % DIM_W; int h = hw / DIM_W;
  int base = h * DIM_W + w;
  const int plane = DIM_H * DIM_W;
  float dx = loc[(4 * a + 0) * plane + base];
  float dy = loc[(4 * a + 1) * plane + base];
  float dw = loc[(4 * a + 2) * plane + base];
  float dh = loc[(4 * a + 3) * plane + base];
  float sc = cls[a * plane + base];
  float x1a = anc[4 * i + 0], y1a = anc[4 * i + 1];
  float x2a = anc[4 * i + 2], y2a = anc[4 * i + 3];
  float aw = x2a - x1a + 1.0f, ah = y2a - y1a + 1.0f;
  float acx = x1a + 0.5f * aw, acy = y1a + 0.5f * ah;
  float cx = dx * aw + acx, cy = dy * ah + acy;
  float ww = expf(dw) * aw, hh = expf(dh) * ah;
  float img_h = ii[0], img_w = ii[1];
  float x1 = fminf(fmaxf(cx - 0.5f * ww, 0.0f), img_w - 1.0f);
  float y1 = fminf(fmaxf(cy - 0.5f * hh, 0.0f), img_h - 1.0f);
  float x2 = fminf(fmaxf(cx + 0.5f * ww, 0.0f), img_w - 1.0f);
  float y2 = fminf(fmaxf(cy + 0.5f * hh, 0.0f), img_h - 1.0f);
  float thr = MIN_FRAC * fmaxf(ii[1], ii[2]);
  bool ok = (x2 - x1 + 1.0f > thr) && (y2 - y1 + 1.0f > thr);
  if (!ok) sc = -INFINITY;
  boxes[4 * i + 0] = x1; boxes[4 * i + 1] = y1;
  boxes[4 * i + 2] = x2; boxes[4 * i + 3] = y2;
  scores[i] = sc;
  atomicAdd(&hist[score_bin(sc)], 1u);
}

// ---------------------------------------------------- 2) top-k cutoff + compaction
__global__ void k_thresh(const u32* __restrict__ hist, u32* __restrict__ misc) {
  if (threadIdx.x == 0 && blockIdx.x == 0) {
    u32 cum = 0; int t = 0;
    for (int b = NBINS - 1; b >= 0; --b) {
      cum += hist[b];
      if (cum >= TOPN) { t = b; break; }
    }
    misc[0] = (u32)t;
  }
}

__global__ void k_compact(const float* __restrict__ scores, u32* __restrict__ misc,
                          u64* __restrict__ cand) {
  int i = blockIdx.x * blockDim.x + threadIdx.x;
  if (i >= N_TOT) return;
  float sc = scores[i];
  if (score_bin(sc) >= (int)misc[0]) {
    u32 slot = atomicAdd(&misc[1], 1u);
    if (slot < CAND_CAP) {
      // sort desc by score, asc by index (matches lax.top_k tie order)
      u64 key = ((u64)flip_f32(sc) << 32) | (u64)(0xFFFFFFFFu - (u32)i);
      cand[slot] = key;
    }
  }
}

// ---------------------------------------------------- 3) bitonic sort (descending, 32768 u64)
__global__ void __launch_bounds__(1024) k_sort_local(u64* cand) {
  __shared__ u64 s[2048];
  int tid = threadIdx.x; int gbase = blockIdx.x * 2048;
  s[tid] = cand[gbase + tid];
  s[tid + 1024] = cand[gbase + tid + 1024];
  for (int k = 2; k <= 2048; k <<= 1) {
    for (int j = k >> 1; j >= 1; j >>= 1) {
      __syncthreads();
      int l = ((tid & ~(j - 1)) << 1) | (tid & (j - 1));
      int p = l | j;
      bool desc = ((gbase + l) & k) == 0;
      u64 a = s[l], b = s[p];
      if (desc ? (a < b) : (a > b)) { s[l] = b; s[p] = a; }
    }
  }
  __syncthreads();
  cand[gbase + tid] = s[tid];
  cand[gbase + tid + 1024] = s[tid + 1024];
}

__global__ void __launch_bounds__(1024) k_sort_global(u64* cand, int j, int k) {
  int t = blockIdx.x * blockDim.x + threadIdx.x;
  if (t >= CAND_CAP / 2) return;
  int l = ((t & ~(j - 1)) << 1) | (t & (j - 1));
  int p = l | j;
  bool desc = (l & k) == 0;
  u64 a = cand[l], b = cand[p];
  if (desc ? (a < b) : (a > b)) { cand[l] = b; cand[p] = a; }
}

__global__ void __launch_bounds__(1024) k_sort_merge_local(u64* cand, int k) {
  __shared__ u64 s[2048];
  int tid = threadIdx.x; int gbase = blockIdx.x * 2048;
  s[tid] = cand[gbase + tid];
  s[tid + 1024] = cand[gbase + tid + 1024];
  for (int j = 1024; j >= 1; j >>= 1) {
    __syncthreads();
    int l = ((tid & ~(j - 1)) << 1) | (tid & (j - 1));
    int p = l | j;
    bool desc = ((gbase + l) & k) == 0;
    u64 a = s[l], b = s[p];
    if (desc ? (a < b) : (a > b)) { s[l] = b; s[p] = a; }
  }
  __syncthreads();
  cand[gbase + tid] = s[tid];
  cand[gbase + tid + 1024] = s[tid + 1024];
}

// ---------------------------------------------------- 4) gather top-12000
__global__ void k_gather(const u64* __restrict__ cand, const float* __restrict__ boxes,
                         const float* __restrict__ scores, float* __restrict__ topb,
                         float* __restrict__ tops) {
  int i = blockIdx.x * blockDim.x + threadIdx.x;
  if (i >= TOPN) return;
  u64 key = cand[i];
  u32 idx = 0xFFFFFFFFu - (u32)(key & 0xFFFFFFFFull);
  if (idx >= N_TOT) idx = 0;   // padding safety (never hit: >=12000 candidates by construction)
  tops[i] = scores[idx];
  float4 b = *(const float4*)(boxes + (size_t)idx * 4);
  *(float4*)(topb + (size_t)i * 4) = b;
}

// ---------------------------------------------------- 5) IoU bitmask
// Block = 256 threads (8 waves) handling 32 rows x all 375 column words.
// Column-box chunks (256 boxes = 4KB) staged via TDM tensor_load_to_lds by wave 0.
__global__ void __launch_bounds__(256) k_mask(const float* __restrict__ topb,
                                              u32* __restrict__ mask) {
  int tid = threadIdx.x;
  int lane = tid & 31;
  int wv = tid >> 5;          // 0..7: column block within chunk
  int rb = blockIdx.x;        // row block 0..374
  int r = rb * 32 + lane;

  __shared__ __align__(16) float scol[256 * 4];   // 4KB column-box chunk

  const float4 rbx = *(const float4*)(topb + (size_t)r * 4);
  float rarea = (rbx.z - rbx.x + 1.0f) * (rbx.w - rbx.y + 1.0f);

  // strictly-lower-triangle words: all j <= i -> zero
  for (int cb = wv; cb < rb; cb += 8) mask[(size_t)r * NWORDS + cb] = 0u;

  for (int ch = rb >> 3; ch < NCHUNK; ++ch) {
    __syncthreads();                       // protect scol from previous readers
    if (wv == 0) {
      u32 ldsa = (u32)(uintptr_t)(&scol[0]);             // low 32b = LDS offset
      u64 ga = (u64)(uintptr_t)(topb + (size_t)ch * 1024);
      tdm_load_4kb(ldsa, ga);                            // gfx1250 TDM DMA
      __builtin_amdgcn_s_wait_tensorcnt(0);
    }
    __syncthreads();
    int cb = ch * 8 + wv;
    if (cb >= rb && cb < NWORDS) {
      const float* cbs = &scol[(wv * 32) * 4];
      u32 word = 0u;
#pragma unroll 4
      for (int c = 0; c < 32; ++c) {
        int j = cb * 32 + c;
        float cx1 = cbs[c * 4 + 0], cy1 = cbs[c * 4 + 1];
        float cx2 = cbs[c * 4 + 2], cy2 = cbs[c * 4 + 3];
        float carea = (cx2 - cx1 + 1.0f) * (cy2 - cy1 + 1.0f);
        float iw = fmaxf(fminf(rbx.z, cx2) - fmaxf(rbx.x, cx1) + 1.0f, 0.0f);
        float ih = fmaxf(fminf(rbx.w, cy2) - fmaxf(rbx.y, cy1) + 1.0f, 0.0f);
        float inter = iw * ih;
        float iou = inter / (rarea + carea - inter);
        if ((iou > IOU_THR) && (j > r)) word |= (1u << c);
      }
      mask[(size_t)r * NWORDS + cb] = word;
    }
  }
}

// ---------------------------------------------------- 6) sequential greedy scan (single wave)
// Scores bulk-staged to LDS with async global->LDS B128 (ASYNCcnt path).
__global__ void __launch_bounds__(32) k_scan(const u32* __restrict__ mask,
                                             const float* __restrict__ tops,
                                             u32* __restrict__ selk, u32* __restrict__ seln,
                                             u32* __restrict__ misc) {
  int lane = threadIdx.x;
  __shared__ u32 rem[NWORDS];
  __shared__ __align__(16) float ssc[TOPN];       // 48KB score cache
  for (int t = lane; t < NWORDS; t += 32) rem[t] = 0u;
  for (int t = lane * 4; t < TOPN; t += 128) {    // 3000 B128 async transfers
    u32 so = (u32)(uintptr_t)(&ssc[t]);
    u64 ga = (u64)(uintptr_t)(tops + t);
    asm volatile("global_load_async_to_lds_b128 %0, %1, off"
                 :: "v"(so), "v"(ga) : "memory");
  }
  asm volatile("s_wait_asynccnt 0" ::: "memory");
  __syncthreads();

  u32 kept = 0, nk = 0;
  for (int i = 0; i < TOPN; ++i) {
    u32 word = rem[i >> 5];                       // uniform LDS broadcast
    bool sup = (word >> (i & 31)) & 1u;
    float sc = ssc[i];
    bool keep = (!sup) && (sc != -INFINITY);
    if (keep) {
      const u32* row = mask + (size_t)i * NWORDS;
      for (int t = lane; t < NWORDS; t += 32) rem[t] |= row[t];
    }
    if (lane == 0) {
      if (keep) { if (kept < POSTN) selk[kept] = (u32)i; kept++; }
      else      { if (nk   < POSTN) seln[nk]   = (u32)i; nk++;  }
    }
    __syncthreads();
  }
  if (lane == 0) { misc[2] = kept; misc[3] = nk; }
}

// ---------------------------------------------------- 7) final selection / output
__global__ void k_out(const float* __restrict__ topb, const float* __restrict__ tops,
                      const u32* __restrict__ selk, const u32* __restrict__ seln,
                      const u32* __restrict__ misc, float* __restrict__ out) {
  int k = blockIdx.x * blockDim.x + threadIdx.x;
  if (k >= POSTN) return;
  u32 kept = misc[2];
  u32 kcap = kept < POSTN ? kept : POSTN;
  u32 i; float flag;
  if ((u32)k < kcap) { i = selk[k]; flag = 1.0f; }
  else               { i = seln[k - kcap]; flag = 0.0f; }
  out[k * 5 + 0] = topb[i * 4 + 0];
  out[k * 5 + 1] = topb[i * 4 + 1];
  out[k * 5 + 2] = topb[i * 4 + 2];
  out[k * 5 + 3] = topb[i * 4 + 3];
  out[k * 5 + 4] = tops[i];
  out[5 * POSTN + k] = flag;    // keep[sel] as 0/1 float
}

// ---------------------------------------------------------------- launch
extern "C" void kernel_launch(void* const* d_in, const int* in_sizes, int n_in,
                              void* d_out, int out_size, void* d_ws, size_t ws_size,
                              hipStream_t stream) {
  (void)in_sizes; (void)n_in; (void)out_size; (void)ws_size;
  const float* cls = (const float*)d_in[0];   // (1,9,128,128)
  const float* loc = (const float*)d_in[1];   // (1,36,128,128)
  const float* anc = (const float*)d_in[2];   // (147456,4)
  const float* ii  = (const float*)d_in[3];   // im_info[3]
  // d_in[4]=topn(12000), d_in[5]=post_topn(1000): compile-time constants here.

  char* ws = (char*)d_ws;
  float* boxes  = (float*)(ws + 0);          // 147456*4 f = 2359296 B
  float* scores = (float*)(ws + 2359296);    // 147456 f   =  589824 B
  u32*  hist    = (u32*) (ws + 2949120);     // 4096 u32   =   16384 B
  u32*  misc    = (u32*) (ws + 2965504);     // 16 u32     =      64 B
  u64*  cand    = (u64*) (ws + 2965568);     // 32768 u64  =  262144 B
  float* topb   = (float*)(ws + 3227712);    // 12000*4 f  =  192000 B
  float* tops   = (float*)(ws + 3419712);    // 12000 f    =   48000 B
  u32*  selk    = (u32*) (ws + 3467712);     // 1000 u32
  u32*  seln    = (u32*) (ws + 3471712);     // 1000 u32
  u32*  mask    = (u32*) (ws + 3475712);     // 12000*375 u32 = 18.0 MB
  float* out = (float*)d_out;

  // zero hist + misc + cand (contiguous region: 278592 B = 69648 u32)
  k_zero<<<(69648 + 255) / 256, 256, 0, stream>>>(hist, 69648);
  k_decode<<<(N_TOT + 255) / 256, 256, 0, stream>>>(cls, loc, anc, ii, boxes, scores, hist);
  k_thresh<<<1, 1, 0, stream>>>(hist, misc);
  k_compact<<<(N_TOT + 255) / 256, 256, 0, stream>>>(scores, misc, cand);

  // bitonic sort, descending, CAND_CAP=32768
  k_sort_local<<<CAND_CAP / 2048, 1024, 0, stream>>>(cand);
  for (int k = 4096; k <= CAND_CAP; k <<= 1) {
    for (int j = k >> 1; j >= 2048; j >>= 1)
      k_sort_global<<<(CAND_CAP / 2 + 1023) / 1024, 1024, 0, stream>>>(cand, j, k);
    k_sort_merge_local<<<CAND_CAP / 2048, 1024, 0, stream>>>(cand, k);
  }

  k_gather<<<(TOPN + 255) / 256, 256, 0, stream>>>(cand, boxes, scores, topb, tops);
  k_mask<<<NWORDS, 256, 0, stream>>>(topb, mask);
  k_scan<<<1, 32, 0, stream>>>(mask, tops, selk, seln, misc);
  k_out<<<(POSTN + 255) / 256, 256, 0, stream>>>(topb, tops, selk, seln, misc, out);
}